// AFAR_34127810134397
// MI455X (gfx1250) — compile-verified
//
#include <hip/hip_runtime.h>
#include <hip/hip_bf16.h>
#include <math.h>

#define CH   256
#define NN   256          // H == W == 256
#define HW   (NN*NN)
#define LDA  257          // padded column stride for Jacobi LDS panel (bank spread)
#define SWEEPS 8
#define EPSV 2.220446049250313e-16f
#define NEG_INF (-3.402823466e38f)

typedef __attribute__((ext_vector_type(2))) float v2f;
typedef __attribute__((ext_vector_type(8))) float v8f;

__device__ __forceinline__ float sigmoidf(float v) { return 1.0f / (1.0f + __expf(-v)); }

// ---------------------------------------------------------------------------
// K1: per-channel mean / max over H*W
// ---------------------------------------------------------------------------
__global__ void k_ch_stats(const float* __restrict__ x,
                           float* __restrict__ mean, float* __restrict__ mx) {
    __shared__ float ss[256], smx[256];
    int c = blockIdx.x, t = threadIdx.x;
    const float* p = x + (size_t)c * HW;
    float s = 0.f, m = NEG_INF;
    for (int i = t; i < HW; i += 256) { float v = p[i]; s += v; m = fmaxf(m, v); }
    ss[t] = s; smx[t] = m; __syncthreads();
    for (int o = 128; o; o >>= 1) {
        if (t < o) { ss[t] += ss[t + o]; smx[t] = fmaxf(smx[t], smx[t + o]); }
        __syncthreads();
    }
    if (t == 0) { mean[c] = ss[0] * (1.0f / HW); mx[c] = smx[0]; }
}

// ---------------------------------------------------------------------------
// K2: shared MLP gate -> sigmoid -> min-max normalize -> per-channel rank k
// ---------------------------------------------------------------------------
__global__ void k_gates(const float* __restrict__ mean, const float* __restrict__ mx,
                        const float* __restrict__ w1, const float* __restrict__ w2,
                        const int* __restrict__ kv, int* __restrict__ kkeep) {
    __shared__ float h[32];                 // [0..15] relu(avg@w1T), [16..31] relu(max@w1T)
    __shared__ float y[256], rmin[256], rmax[256];
    int t = threadIdx.x;
    if (t < 32) {
        const float* v = (t < 16) ? mean : mx;
        int j = t & 15;
        float acc = 0.f;
        for (int c = 0; c < 256; c++) acc += v[c] * w1[j * 256 + c];
        h[t] = fmaxf(acc, 0.f);
    }
    __syncthreads();
    float acc = 0.f;
    for (int j = 0; j < 16; j++) acc += (h[j] + h[16 + j]) * w2[t * 16 + j];
    float yv = sigmoidf(acc);
    y[t] = yv; rmin[t] = yv; rmax[t] = yv; __syncthreads();
    for (int o = 128; o; o >>= 1) {
        if (t < o) { rmin[t] = fminf(rmin[t], rmin[t + o]); rmax[t] = fmaxf(rmax[t], rmax[t + o]); }
        __syncthreads();
    }
    float ycn = (y[t] - rmin[0]) / (rmax[0] - rmin[0] + 1e-20f);
    int k = (int)floorf(256.0f * (float)kv[0] * ycn);
    kkeep[t] = max(0, min(256, k));
}

// ---------------------------------------------------------------------------
// K0: init fused channel-reduction accumulators
// ---------------------------------------------------------------------------
__global__ void k_init(float* __restrict__ sumb, float* __restrict__ maxb) {
    int i = blockIdx.x * blockDim.x + threadIdx.x;
    if (i < HW) { sumb[i] = 0.f; maxb[i] = NEG_INF; }
}

// ---------------------------------------------------------------------------
// K4: one-sided Hestenes-Jacobi SVD per channel, panel fully LDS-resident.
//     Output: Ubuf[ch][i*256+r] = masked/normalized left singular vectors
//             (u_i[r] / 1 for kept ranks, 0 for dropped -> U_m, column-major)
// ---------------------------------------------------------------------------
__global__ void __launch_bounds__(1024)
k_jacobi(const float* __restrict__ x, const int* __restrict__ kkeep,
         float* __restrict__ Ubuf) {
    extern __shared__ float lds[];
    float* B    = lds;                   // 256*LDA  column-major panel
    float* cs_c = lds + 256 * LDA;       // 128
    float* cs_s = cs_c + 128;            // 128
    float* s2   = cs_s + 128;            // 256 squared singular values
    float* isv  = s2 + 256;              // 256 masked 1/s_i

    int ch = blockIdx.x, tid = threadIdx.x;
    const float* A = x + (size_t)ch * HW;

    // B[j*LDA + r] = A[r][j]  (coalesced global reads over j)
    for (int idx = tid; idx < HW; idx += 1024) {
        int r = idx >> 8, j = idx & 255;
        B[j * LDA + r] = A[idx];
    }
    __syncthreads();

    int pp = tid >> 3, sub = tid & 7;    // 128 pairs x 8 threads
    for (int sw = 0; sw < SWEEPS; sw++) {
        for (int rd = 0; rd < 255; rd++) {
            int p, q;                    // round-robin disjoint pairing
            if (pp == 0) { p = 255; q = rd % 255; }
            else { p = (rd + pp) % 255; q = (rd + 255 - pp) % 255; }
            const float* bp = B + p * LDA;
            const float* bq = B + q * LDA;
            float dpp = 0.f, dqq = 0.f, dpq = 0.f;
            for (int r = sub; r < 256; r += 8) {
                float a = bp[r], b = bq[r];
                dpp += a * a; dqq += b * b; dpq += a * b;
            }
            for (int o = 4; o; o >>= 1) {
                dpp += __shfl_xor(dpp, o, 8);
                dqq += __shfl_xor(dqq, o, 8);
                dpq += __shfl_xor(dpq, o, 8);
            }
            if (sub == 0) {
                float c = 1.f, s = 0.f;
                if (dpq * dpq > 1e-26f * dpp * dqq) {
                    float tau = (dqq - dpp) / (2.f * dpq);
                    float tt  = (tau >= 0.f ? 1.f : -1.f) /
                                (fabsf(tau) + sqrtf(1.f + tau * tau));
                    c = 1.0f / sqrtf(1.f + tt * tt);
                    s = tt * c;
                }
                cs_c[pp] = c; cs_s[pp] = s;
            }
            __syncthreads();
            float c = cs_c[pp], s = cs_s[pp];
            float* wp = B + p * LDA;
            float* wq = B + q * LDA;
            for (int r = sub; r < 256; r += 8) {
                float a = wp[r], b = wq[r];
                wp[r] = c * a - s * b;
                wq[r] = s * a + c * b;
            }
            __syncthreads();
        }
    }

    // squared singular values = column norms
    if (tid < 256) {
        const float* bc = B + tid * LDA;
        float acc = 0.f;
        for (int r = 0; r < 256; r++) { float v = bc[r]; acc += v * v; }
        s2[tid] = acc;
    }
    __syncthreads();
    // rank by descending s (tie-break on index), mask top-k, prep 1/s
    if (tid < 256) {
        float mine = s2[tid];
        int rank = 0;
        for (int j = 0; j < 256; j++) {
            float o = s2[j];
            rank += (o > mine) || (o == mine && j < tid);
        }
        int keep = rank < kkeep[ch];
        isv[tid] = (keep && mine > 0.f) ? rsqrtf(mine) : 0.f;
    }
    __syncthreads();
    float* Uc = Ubuf + (size_t)ch * HW;  // column-major: Uc[i*256 + r]
    for (int idx = tid; idx < HW; idx += 1024) {
        int i = idx >> 8, r = idx & 255;
        Uc[idx] = B[i * LDA + r] * isv[i];
    }
}

// ---------------------------------------------------------------------------
// K5: recon_c = U_m (U_m^T A_c) via native fp32 WMMA (16x16x4), T in LDS.
//     Fused channel mean/max via float atomics (no 64MB x2 intermediate).
// ---------------------------------------------------------------------------
__global__ void __launch_bounds__(1024)
k_recon(const float* __restrict__ x, const float* __restrict__ Ubuf,
        float* __restrict__ sumb, float* __restrict__ maxb) {
    extern __shared__ float lds[];
    float* T = lds;                       // 256*256 intermediate, row-major
    int ch = blockIdx.x;
    int tid = threadIdx.x, wid = tid >> 5, lane = tid & 31;
    const float* A  = x    + (size_t)ch * HW;
    const float* Uc = Ubuf + (size_t)ch * HW;   // Uc[i*256+r] = U[r][i]
    int half = lane >> 4;
    int l15  = lane & 15;
    int kb   = half ? 2 : 0;              // K-split across lane halves (A/B frags)

    // Phase 1: T[i][j] = sum_r U[r][i] * A[r][j]   (T = U_m^T A)
    for (int s = 0; s < 8; s++) {
        int tt = wid * 8 + s;             // 256 tiles / 32 waves
        int i0 = (tt >> 4) * 16;
        int j0 = (tt & 15) * 16;
        v8f acc = {};
        const float* ua = Uc + (i0 + l15) * 256;   // row of U^T, contiguous in r
        for (int r0 = 0; r0 < 256; r0 += 4) {
            v2f av = { ua[r0 + kb], ua[r0 + kb + 1] };
            v2f bv = { A[(r0 + kb) * 256 + j0 + l15],
                       A[(r0 + kb + 1) * 256 + j0 + l15] };
            acc = __builtin_amdgcn_wmma_f32_16x16x4_f32(
                      false, av, false, bv, (short)0, acc, false, false);
        }
        int row0 = i0 + (half ? 8 : 0);
        int col  = j0 + l15;
#pragma unroll
        for (int r = 0; r < 8; r++) T[(row0 + r) * 256 + col] = acc[r];
    }
    __syncthreads();

    // Phase 2: R[m][j] = sum_i U[m][i] * T[i][j]; fuse mean/max over channels
    for (int s = 0; s < 8; s++) {
        int tt = wid * 8 + s;
        int m0 = (tt >> 4) * 16;
        int n0 = (tt & 15) * 16;
        v8f acc = {};
        for (int i0 = 0; i0 < 256; i0 += 4) {
            v2f av = { Uc[(i0 + kb) * 256 + m0 + l15],
                       Uc[(i0 + kb + 1) * 256 + m0 + l15] };
            v2f bv = { T[(i0 + kb) * 256 + n0 + l15],
                       T[(i0 + kb + 1) * 256 + n0 + l15] };
            acc = __builtin_amdgcn_wmma_f32_16x16x4_f32(
                      false, av, false, bv, (short)0, acc, false, false);
        }
        int row0 = m0 + (half ? 8 : 0);
        int col  = n0 + l15;
#pragma unroll
        for (int r = 0; r < 8; r++) {
            float v = acc[r];
            int px = (row0 + r) * 256 + col;
            __hip_atomic_fetch_add(&sumb[px], v, __ATOMIC_RELAXED, __HIP_MEMORY_SCOPE_AGENT);
            __hip_atomic_fetch_max(&maxb[px], v, __ATOMIC_RELAXED, __HIP_MEMORY_SCOPE_AGENT);
        }
    }
}

// ---------------------------------------------------------------------------
// K6: 7x7 spatial-attention conv on [avg;max], clamp to EPS, sigmoid
// ---------------------------------------------------------------------------
__global__ void k_spatial(const float* __restrict__ sumb, const float* __restrict__ maxb,
                          const float* __restrict__ cw, float* __restrict__ out) {
    int h = blockIdx.x, w = threadIdx.x;
    float acc = 0.f;
#pragma unroll
    for (int kh = 0; kh < 7; kh++) {
        int hh = h + kh - 3;
        if ((unsigned)hh >= 256u) continue;
#pragma unroll
        for (int kw = 0; kw < 7; kw++) {
            int ww = w + kw - 3;
            if ((unsigned)ww >= 256u) continue;
            int px = hh * 256 + ww;
            acc += cw[kh * 7 + kw] * (sumb[px] * (1.0f / 256.f))
                 + cw[49 + kh * 7 + kw] * maxb[px];
        }
    }
    if (acc < EPSV) acc = EPSV;
    out[h * 256 + w] = sigmoidf(acc);
}

// ---------------------------------------------------------------------------
extern "C" void kernel_launch(void* const* d_in, const int* in_sizes, int n_in,
                              void* d_out, int out_size, void* d_ws, size_t ws_size,
                              hipStream_t stream) {
    (void)in_sizes; (void)n_in; (void)out_size; (void)ws_size;
    const float* x  = (const float*)d_in[0];
    const float* w1 = (const float*)d_in[1];
    const float* w2 = (const float*)d_in[2];
    const float* cw = (const float*)d_in[3];
    const int*   kv = (const int*)d_in[4];
    float* out = (float*)d_out;

    float* ws      = (float*)d_ws;
    float* ch_mean = ws;                  // 256
    float* ch_max  = ws + 256;            // 256
    int*   kkeep   = (int*)(ws + 512);    // 256 (padded to 1024 floats total)
    float* sumb    = ws + 1024;           // 65536
    float* maxb    = sumb + HW;           // 65536
    float* Ubuf    = maxb + HW;           // 256*256*256 = 64 MB

    k_ch_stats<<<256, 256, 0, stream>>>(x, ch_mean, ch_max);
    k_gates<<<1, 256, 0, stream>>>(ch_mean, ch_max, w1, w2, kv, kkeep);
    k_init<<<HW / 256, 256, 0, stream>>>(sumb, maxb);

    size_t jac_lds = (size_t)(256 * LDA + 128 + 128 + 256 + 256) * sizeof(float);
    (void)hipFuncSetAttribute((const void*)k_jacobi,
                              hipFuncAttributeMaxDynamicSharedMemorySize, (int)jac_lds);
    k_jacobi<<<256, 1024, jac_lds, stream>>>(x, kkeep, Ubuf);

    size_t gemm_lds = (size_t)HW * sizeof(float);
    (void)hipFuncSetAttribute((const void*)k_recon,
                              hipFuncAttributeMaxDynamicSharedMemorySize, (int)gemm_lds);
    k_recon<<<256, 1024, gemm_lds, stream>>>(x, Ubuf, sumb, maxb);

    k_spatial<<<256, 256, 0, stream>>>(sumb, maxb, cw, out);
}